// MultiDiceLoss_59442347376953
// MI455X (gfx1250) — compile-verified
//
#include <hip/hip_runtime.h>
#include <hip/hip_bf16.h>

// ---------------------------------------------------------------------------
// MultiDiceLoss for MI455X (gfx1250, wave32).
// Memory-bound streaming reduction: 160 MiB single pass -> ~7.2 us at 23.3 TB/s.
// Strategy: b128 non-temporal loads, register accumulation, wave32 reduction
// done with V_WMMA_F32_16X16X4_F32 (B = all-ones so only the documented A and
// C/D layouts matter), one atomicAdd per wave into tiny scratch, tiny epilogue.
// ---------------------------------------------------------------------------

typedef float v2f   __attribute__((ext_vector_type(2)));
typedef float v8f   __attribute__((ext_vector_type(8)));
typedef float f32x4 __attribute__((ext_vector_type(4)));
typedef int   s32x4 __attribute__((ext_vector_type(4)));

#define HW        (1024 * 1024)   // pixels per image
#define NIMG      8
#define NCLS      4
#define BPI       128             // blocks per image
#define TPB       256             // 8 wave32 per block
#define PIXPB     (HW / BPI)      // 8192 pixels per block
#define ITERS     (PIXPB / (TPB * 4))  // 8 float4-iterations per thread
#define SMOOTH    1e-5f

// Full wave32 sum using two WMMA f32 16x16x4 ops with B == all-ones.
// WMMA #1: A slots (lane l<16 -> A[l][0],A[l][1]; lane l>=16 -> A[l-16][2],A[l-16][3])
// both set to x  =>  D[m][n] = 2*(x[m] + x[m+16]) for every column n.
// Summing the 8 D VGPRs gives s_half (uniform per lane-half); WMMA #2 on s_half
// gives 2*(s_low + s_high) = 4 * Sum(x over 32 lanes) in every lane/register.
__device__ __forceinline__ float wave_reduce_sum_wmma(float x) {
  v2f ones = {1.0f, 1.0f};
  v8f zero = {};
  v2f a = {x, x};
  v8f d = __builtin_amdgcn_wmma_f32_16x16x4_f32(false, a, false, ones,
                                                (short)0, zero, false, false);
  float y = ((d[0] + d[1]) + (d[2] + d[3])) + ((d[4] + d[5]) + (d[6] + d[7]));
  v2f a2 = {y, y};
  v8f d2 = __builtin_amdgcn_wmma_f32_16x16x4_f32(false, a2, false, ones,
                                                 (short)0, zero, false, false);
  return d2[0] * 0.25f;
}

__global__ void zero_ws_kernel(float* ws) {
  int i = threadIdx.x;
  if (i < NIMG * 12) ws[i] = 0.0f;
}

__global__ __launch_bounds__(TPB) void dice_partial_kernel(
    const float* __restrict__ input,   // [N, C, H, W]
    const int*   __restrict__ target,  // [N, H, W]
    float*       __restrict__ ws) {    // [N][12]: sum_p[4], inter[4], cnt[4]
  const int n = blockIdx.y;
  const f32x4* plane = (const f32x4*)(input + (size_t)n * NCLS * HW);
  const s32x4* tgt   = (const s32x4*)(target + (size_t)n * HW);
  const int HW4 = HW / 4;

  int idx = blockIdx.x * (PIXPB / 4) + threadIdx.x;  // in float4 units

  float sp0 = 0.f, sp1 = 0.f, sp2 = 0.f, sp3 = 0.f;  // sum of probs
  float it0 = 0.f, it1 = 0.f, it2 = 0.f, it3 = 0.f;  // intersection
  float ct0 = 0.f, ct1 = 0.f, ct2 = 0.f, ct3 = 0.f;  // onehot counts

#pragma unroll 2
  for (int i = 0; i < ITERS; ++i, idx += TPB) {
    // one b128 per class plane + one b128 of targets, non-temporal (streaming)
    f32x4 l0 = __builtin_nontemporal_load(plane + 0 * HW4 + idx);
    f32x4 l1 = __builtin_nontemporal_load(plane + 1 * HW4 + idx);
    f32x4 l2 = __builtin_nontemporal_load(plane + 2 * HW4 + idx);
    f32x4 l3 = __builtin_nontemporal_load(plane + 3 * HW4 + idx);
    s32x4 tv = __builtin_nontemporal_load(tgt + idx);

#pragma unroll
    for (int j = 0; j < 4; ++j) {
      float x0 = l0[j], x1 = l1[j], x2 = l2[j], x3 = l3[j];
      int   t  = tv[j];
      float m  = fmaxf(fmaxf(x0, x1), fmaxf(x2, x3));
      float e0 = __expf(x0 - m);
      float e1 = __expf(x1 - m);
      float e2 = __expf(x2 - m);
      float e3 = __expf(x3 - m);
      float r  = __builtin_amdgcn_rcpf((e0 + e1) + (e2 + e3));
      float p0 = e0 * r, p1 = e1 * r, p2 = e2 * r, p3 = e3 * r;
      sp0 += p0; sp1 += p1; sp2 += p2; sp3 += p3;
      it0 += (t == 0) ? p0 : 0.0f;  ct0 += (t == 0) ? 1.0f : 0.0f;
      it1 += (t == 1) ? p1 : 0.0f;  ct1 += (t == 1) ? 1.0f : 0.0f;
      it2 += (t == 2) ? p2 : 0.0f;  ct2 += (t == 2) ? 1.0f : 0.0f;
      it3 += (t == 3) ? p3 : 0.0f;  ct3 += (t == 3) ? 1.0f : 0.0f;
    }
  }

  // Uniform control flow: every lane of every wave executes these (EXEC all 1s,
  // as WMMA requires). Each call leaves the wave-total in every lane.
  float r0  = wave_reduce_sum_wmma(sp0);
  float r1  = wave_reduce_sum_wmma(sp1);
  float r2  = wave_reduce_sum_wmma(sp2);
  float r3  = wave_reduce_sum_wmma(sp3);
  float r4  = wave_reduce_sum_wmma(it0);
  float r5  = wave_reduce_sum_wmma(it1);
  float r6  = wave_reduce_sum_wmma(it2);
  float r7  = wave_reduce_sum_wmma(it3);
  float r8  = wave_reduce_sum_wmma(ct0);
  float r9  = wave_reduce_sum_wmma(ct1);
  float r10 = wave_reduce_sum_wmma(ct2);
  float r11 = wave_reduce_sum_wmma(ct3);

  if ((threadIdx.x & 31u) == 0u) {  // one lane per wave32
    float* dst = ws + n * 12;
    atomicAdd(dst + 0,  r0);  atomicAdd(dst + 1,  r1);
    atomicAdd(dst + 2,  r2);  atomicAdd(dst + 3,  r3);
    atomicAdd(dst + 4,  r4);  atomicAdd(dst + 5,  r5);
    atomicAdd(dst + 6,  r6);  atomicAdd(dst + 7,  r7);
    atomicAdd(dst + 8,  r8);  atomicAdd(dst + 9,  r9);
    atomicAdd(dst + 10, r10); atomicAdd(dst + 11, r11);
  }
}

__global__ void dice_finalize_kernel(const float* __restrict__ ws,
                                     const float* __restrict__ weights,
                                     float* __restrict__ out) {
  if (threadIdx.x == 0) {
    float total = 0.0f;
    for (int c = 0; c < NCLS; ++c) {
      float acc = 0.0f;
      for (int n = 0; n < NIMG; ++n) {
        float sp = ws[n * 12 + c];
        float it = ws[n * 12 + 4 + c];
        float ct = ws[n * 12 + 8 + c];
        float dice = (2.0f * it + SMOOTH) / (sp + ct + SMOOTH);
        acc += dice;
      }
      total += (1.0f - acc * (1.0f / NIMG)) * weights[c];
    }
    out[0] = total;
  }
}

extern "C" void kernel_launch(void* const* d_in, const int* in_sizes, int n_in,
                              void* d_out, int out_size, void* d_ws, size_t ws_size,
                              hipStream_t stream) {
  (void)in_sizes; (void)n_in; (void)out_size; (void)ws_size;
  const float* input   = (const float*)d_in[0];
  const int*   target  = (const int*)d_in[1];
  const float* weights = (const float*)d_in[2];
  float*       out     = (float*)d_out;
  float*       ws      = (float*)d_ws;

  zero_ws_kernel<<<1, 128, 0, stream>>>(ws);
  dim3 grid(BPI, NIMG);
  dice_partial_kernel<<<grid, TPB, 0, stream>>>(input, target, ws);
  dice_finalize_kernel<<<1, 32, 0, stream>>>(ws, weights, out);
}